// BitFlipLayer_20444044329820
// MI455X (gfx1250) — compile-verified
//
#include <hip/hip_runtime.h>
#include <stdint.h>

// Native clang vector types: accepted by __builtin_nontemporal_load/store and
// lower to single global_load_b128/global_store_b128 per lane on gfx1250.
typedef unsigned int u32x4 __attribute__((ext_vector_type(4)));
typedef int          i32x4 __attribute__((ext_vector_type(4)));

// ---------------------------------------------------------------------------
// Vectorized main kernel: each lane processes 4 elements (16 B per stream).
// Streaming op, zero reuse, working set >> 192 MB L2  ->  non-temporal hints.
// ---------------------------------------------------------------------------
__global__ __launch_bounds__(256) void bitflip_vec4_kernel(
    const u32x4* __restrict__ xbits,
    const i32x4* __restrict__ bit_pos,
    const i32x4* __restrict__ flip_mask,
    u32x4* __restrict__       out,
    int nvec)
{
    const int i = blockIdx.x * blockDim.x + threadIdx.x;
    if (i >= nvec) return;

    // th:NT loads — stream through, don't pollute L2 (one-shot data).
    u32x4 b = __builtin_nontemporal_load(&xbits[i]);
    i32x4 p = __builtin_nontemporal_load(&bit_pos[i]);
    i32x4 m = __builtin_nontemporal_load(&flip_mask[i]);

    // sel = 1 where flip_mask != 0 (ext-vector compare yields -1/0 -> mask to 1/0)
    i32x4 sel = (m != 0) & 1;
    // reference: bit index 0 is the MSB -> flip bit (31 - bit_pos)
    u32x4 sh  = (u32x4)(31 - (p & 31));
    u32x4 r   = b ^ (((u32x4)sel) << sh);

    // th:NT store — result is not re-read by this dispatch.
    __builtin_nontemporal_store(r, &out[i]);
}

// ---------------------------------------------------------------------------
// Scalar tail for n % 4 (zero for the 32x1024x1024 shape; kept for generality).
// ---------------------------------------------------------------------------
__global__ void bitflip_tail_kernel(
    const unsigned int* __restrict__ xbits,
    const int* __restrict__          bit_pos,
    const int* __restrict__          flip_mask,
    unsigned int* __restrict__       out,
    int start, int n)
{
    const int i = start + (int)threadIdx.x;
    if (i < n) {
        unsigned int sel = (flip_mask[i] != 0) ? 1u : 0u;
        unsigned int sh  = 31u - ((unsigned int)bit_pos[i] & 31u);
        out[i] = xbits[i] ^ (sel << sh);
    }
}

extern "C" void kernel_launch(void* const* d_in, const int* in_sizes, int n_in,
                              void* d_out, int out_size, void* d_ws, size_t ws_size,
                              hipStream_t stream)
{
    (void)n_in; (void)d_ws; (void)ws_size; (void)out_size;

    // setup_inputs() order: x (f32), bit_pos (i32), flip_mask (i32)
    const unsigned int* xbits = (const unsigned int*)d_in[0]; // f32 reinterpreted as bits
    const int*          bpos  = (const int*)d_in[1];
    const int*          mask  = (const int*)d_in[2];
    unsigned int*       out   = (unsigned int*)d_out;         // f32 output as bits

    const int n    = in_sizes[0];
    const int nvec = n >> 2;       // 8,388,608 for 32*1024*1024
    const int rem  = n & 3;

    if (nvec > 0) {
        const int threads = 256;                       // 8 wave32 per block
        const int blocks  = (nvec + threads - 1) / threads; // 32768 blocks
        bitflip_vec4_kernel<<<blocks, threads, 0, stream>>>(
            (const u32x4*)xbits, (const i32x4*)bpos, (const i32x4*)mask,
            (u32x4*)out, nvec);
    }
    if (rem > 0) {
        bitflip_tail_kernel<<<1, 4, 0, stream>>>(
            xbits, bpos, mask, out, nvec << 2, n);
    }
}